// WSGCN_17600775979420
// MI455X (gfx1250) — compile-verified
//
#include <hip/hip_runtime.h>
#include <math.h>

// ---------------------------------------------------------------------------
// WSGCN on MI455X (gfx1250, wave32).
// Edge phase: atomics into L2 (node state fits in 192MB L2).
// Layer-2 GEMM (n x 96) @ (96 x 32): V_WMMA_F32_16X16X4_F32 chains.
// Head: one 128-thread block per 81-node board, staged in LDS.
// ---------------------------------------------------------------------------

#define N_NODES_C (81 * 4096)   // 331776, divisible by 16 and 128
#define HIDC 32

typedef __attribute__((ext_vector_type(2))) float v2f;
typedef __attribute__((ext_vector_type(8))) float v8f;

// ---------------- init ----------------
__global__ void k_init_nodes(float* deg, float* mp, float* mn, float* sp, float* sn,
                             float* hp1, float* hn1, int n) {
  int i = blockIdx.x * blockDim.x + threadIdx.x;
  if (i >= n) return;
  deg[i] = 0.f;
  mp[i] = -1e30f;
  mn[i] = -1e30f;
  sp[i] = 0.f;
  sn[i] = 0.f;
  hp1[i] = 0.f;
  hn1[i] = 0.f;
}

__global__ void k_zero_feat(float* a, float* b, int cnt) {
  int i = blockIdx.x * blockDim.x + threadIdx.x;
  if (i >= cnt) return;
  a[i] = 0.f;
  b[i] = 0.f;
}

// ---------------- edge softmax: segment max ----------------
__global__ void k_edge_max(const float* __restrict__ w, const int* __restrict__ dst,
                           float* mp, float* mn, int E) {
  int e = blockIdx.x * blockDim.x + threadIdx.x;
  if (e >= E) return;
  float wv = w[e];
  int d = dst[e];
  // all candidate values are > 0, init is -1e30 -> signed-int compare of float
  // bit patterns is monotonic for this case.
  if (wv > 0.f) {
    atomicMax((int*)(mp + d), __float_as_int(wv));
  } else if (wv < 0.f) {
    atomicMax((int*)(mn + d), __float_as_int(-wv));
  }
}

// ---------------- edge softmax: exp + segment sum (+ degree) ----------------
__global__ void k_edge_expsum(const float* __restrict__ w, const int* __restrict__ dst,
                              const float* __restrict__ mp, const float* __restrict__ mn,
                              float* sp, float* sn, float* coef, float* deg, int E) {
  int e = blockIdx.x * blockDim.x + threadIdx.x;
  if (e >= E) return;
  float wv = w[e];
  int d = dst[e];
  atomicAdd(deg + d, 1.0f);  // deg = segment_sum(ones)
  if (wv > 0.f) {
    float ee = expf(wv - mp[d]);
    coef[e] = ee;
    atomicAdd(sp + d, ee);
  } else if (wv < 0.f) {
    float ee = expf(-wv - mn[d]);
    coef[e] = ee;
    atomicAdd(sn + d, ee);
  } else {
    coef[e] = 0.f;
  }
}

// ------- normalize coef (store signed), layer-1 aggregation (h0 = deg) -------
__global__ void k_edge_norm_agg1(const float* __restrict__ w, const int* __restrict__ src,
                                 const int* __restrict__ dst,
                                 const float* __restrict__ sp, const float* __restrict__ sn,
                                 const float* __restrict__ deg,
                                 float* coef, float* hp1, float* hn1, int E) {
  int e = blockIdx.x * blockDim.x + threadIdx.x;
  if (e >= E) return;
  float wv = w[e];
  int d = dst[e];
  if (wv > 0.f) {
    float a = coef[e] / fmaxf(sp[d], 1e-20f);
    coef[e] = a;  // positive coefficient
    atomicAdd(hp1 + d, a * deg[src[e]]);
  } else if (wv < 0.f) {
    float a = coef[e] / fmaxf(sn[d], 1e-20f);
    coef[e] = -a;  // negative marks "negative branch"
    atomicAdd(hn1 + d, a * deg[src[e]]);
  }
}

// ---------------- layer-1 node update: relu(comb3 @ W1^T + b1 + bias1) -------
__global__ void k_node_l1(const float* __restrict__ deg, const float* __restrict__ hp1,
                          const float* __restrict__ hn1,
                          const float* __restrict__ W1, const float* __restrict__ b1,
                          const float* __restrict__ bias1,
                          const float* c1s, const float* c1p, const float* c1n,
                          float* __restrict__ h1, int n) {
  int t = blockIdx.x * blockDim.x + threadIdx.x;
  if (t >= n * HIDC) return;
  int i = t >> 5;
  int j = t & 31;
  float c0 = c1s[0] * deg[i];
  float cp = c1p[0] * hp1[i];
  float cn = c1n[0] * hn1[i];
  float v = W1[j * 3 + 0] * c0 + W1[j * 3 + 1] * cp + W1[j * 3 + 2] * cn + b1[j] + bias1[j];
  h1[t] = v > 0.f ? v : 0.f;
}

// ---------------- layer-2 aggregation: lane = feature, wave covers 1 edge ----
__global__ void k_agg2(const int* __restrict__ src, const int* __restrict__ dst,
                       const float* __restrict__ coef, const float* __restrict__ h1,
                       float* hp2, float* hn2, int E) {
  int t = blockIdx.x * blockDim.x + threadIdx.x;
  int e = t >> 5;
  if (e >= E) return;
  int lane = t & 31;
  float c = coef[e];
  if (c == 0.f) return;
  int s = src[e];
  int d = dst[e];
  float v = h1[s * HIDC + lane];  // coalesced 128B per edge
  if (c > 0.f) {
    atomicAdd(hp2 + d * HIDC + lane, c * v);
  } else {
    atomicAdd(hn2 + d * HIDC + lane, (-c) * v);
  }
}

// --------- layer-2 GEMM via WMMA f32 16x16x4: (n x 96) @ (96 x 32) ----------
// One wave per 16 nodes; two 16-col output tiles; K=96 -> 24 WMMA per tile.
__global__ void k_gemm_l2(const float* __restrict__ h1, const float* __restrict__ hp2,
                          const float* __restrict__ hn2, const float* __restrict__ W2,
                          const float* __restrict__ b2, const float* __restrict__ bias2,
                          const float* c2s_p, const float* c2p_p, const float* c2n_p,
                          float* __restrict__ h2, int n) {
  const int wave = (blockIdx.x * blockDim.x + threadIdx.x) >> 5;
  const int lane = threadIdx.x & 31;
  const int m0 = wave * 16;
  if (m0 >= n) return;  // uniform per-wave (n % 16 == 0), EXEC stays all-ones
  const int half = lane >> 4;  // 0: K = k0,k0+1   1: K = k0+2,k0+3
  const int lr = lane & 15;

  const float scl[3] = {c2s_p[0], c2p_p[0], c2n_p[0]};
  const float* rows[3] = {h1 + (size_t)(m0 + lr) * HIDC,
                          hp2 + (size_t)(m0 + lr) * HIDC,
                          hn2 + (size_t)(m0 + lr) * HIDC};
  const float* w2r0 = W2 + (size_t)(0 + lr) * 96;   // B cols j0=0..15
  const float* w2r1 = W2 + (size_t)(16 + lr) * 96;  // B cols j0=16..31

  v8f c0 = {};
  v8f c1 = {};
#pragma unroll
  for (int k0 = 0; k0 < 96; k0 += 4) {
    const int ka = k0 + half * 2;  // even, never crosses a 32-feature segment
    const int seg = ka >> 5;
    const int kk = ka & 31;
    v2f a;
    a.x = scl[seg] * rows[seg][kk];
    a.y = scl[seg] * rows[seg][kk + 1];
    v2f b0;
    b0.x = w2r0[ka];
    b0.y = w2r0[ka + 1];
    v2f b1;
    b1.x = w2r1[ka];
    b1.y = w2r1[ka + 1];
    c0 = __builtin_amdgcn_wmma_f32_16x16x4_f32(false, a, false, b0, (short)0, c0, false, false);
    c1 = __builtin_amdgcn_wmma_f32_16x16x4_f32(false, a, false, b1, (short)0, c1, false, false);
  }
  // C/D layout: VGPR v, lanes 0-15 -> M=v, N=lane ; lanes 16-31 -> M=v+8, N=lane-16
#pragma unroll
  for (int v = 0; v < 8; v++) {
    const int row = m0 + v + half * 8;
    const int col = lr;
    float x0 = c0[v] + b2[col] + bias2[col];
    float x1 = c1[v] + b2[16 + col] + bias2[16 + col];
    h2[(size_t)row * HIDC + col] = x0 > 0.f ? x0 : 0.f;
    h2[(size_t)row * HIDC + 16 + col] = x1 > 0.f ? x1 : 0.f;
  }
}

// ---------------- head: conv1(k=64,s=64) -> pool2 -> conv2(k=3,s=3) -> FCs ---
__global__ __launch_bounds__(128) void k_head(
    const float* __restrict__ h1, const float* __restrict__ h2,
    const float* __restrict__ k1, const float* __restrict__ bk1,
    const float* __restrict__ k2, const float* __restrict__ bk2,
    const float* __restrict__ Wfc, const float* __restrict__ bfc,
    const float* __restrict__ Wc, const float* __restrict__ bc,
    float* __restrict__ out) {
  const int b = blockIdx.x;
  const int tid = threadIdx.x;

  __shared__ float xfeat[80 * 64];  // node features for positions 0..79
  __shared__ float k1s[16 * 64];
  __shared__ float convv[16 * 80];
  __shared__ float poolv[16 * 40];
  __shared__ float yv[416];
  __shared__ float zv[128];
  __shared__ float logits[10];
  __shared__ float red[2];

  // stage gather: x[b,0,p*64+f] = concat(h1,h2)[b*81+p, f]
  for (int i = tid; i < 80 * 64; i += 128) {
    int p = i >> 6;
    int f = i & 63;
    int node = b * 81 + p;
    xfeat[i] = (f < 32) ? h1[(size_t)node * 32 + f] : h2[(size_t)node * 32 + (f - 32)];
  }
  for (int i = tid; i < 16 * 64; i += 128) k1s[i] = k1[i];
  __syncthreads();

  // conv1 (only positions 0..79 survive the [:80] slice)
  for (int idx = tid; idx < 16 * 80; idx += 128) {
    int oc = idx / 80;
    int p = idx % 80;
    float s = bk1[oc];
    const float* kk = k1s + oc * 64;
    const float* xx = xfeat + p * 64;
#pragma unroll 8
    for (int t = 0; t < 64; t++) s += kk[t] * xx[t];
    convv[idx] = s;
  }
  __syncthreads();

  // max-pool pairs -> (16,40)
  for (int idx = tid; idx < 16 * 40; idx += 128) {
    int ic = idx / 40;
    int q = idx % 40;
    poolv[idx] = fmaxf(convv[ic * 80 + 2 * q], convv[ic * 80 + 2 * q + 1]);
  }
  __syncthreads();

  // conv2: (32,16,3) stride 3 -> (32,13); flatten index = oc*13+q
  for (int idx = tid; idx < 32 * 13; idx += 128) {
    int oc = idx / 13;
    int q = idx % 13;
    float s = bk2[oc];
#pragma unroll
    for (int ic = 0; ic < 16; ic++) {
      const float* kk = k2 + (oc * 16 + ic) * 3;
      const float* pp = poolv + ic * 40 + 3 * q;
      s += kk[0] * pp[0] + kk[1] * pp[1] + kk[2] * pp[2];
    }
    yv[idx] = s;
  }
  __syncthreads();

  // fc1 -> relu (128 outputs, one per thread)
  {
    float s = bfc[tid];
    const float* wr = Wfc + (size_t)tid * 416;
#pragma unroll 8
    for (int i = 0; i < 416; i++) s += wr[i] * yv[i];
    zv[tid] = s > 0.f ? s : 0.f;
  }
  __syncthreads();

  // fc2 (10 logits)
  if (tid < 10) {
    float s = bc[tid];
    const float* wr = Wc + tid * 128;
#pragma unroll 8
    for (int j = 0; j < 128; j++) s += wr[j] * zv[j];
    logits[tid] = s;
  }
  __syncthreads();

  if (tid == 0) {
    float m = logits[0];
    for (int c = 1; c < 10; c++) m = fmaxf(m, logits[c]);
    float se = 0.f;
    for (int c = 0; c < 10; c++) se += expf(logits[c] - m);
    red[0] = m;
    red[1] = logf(se);
  }
  __syncthreads();

  if (tid < 10) out[(size_t)b * 10 + tid] = logits[tid] - red[0] - red[1];
}

// ---------------------------------------------------------------------------
extern "C" void kernel_launch(void* const* d_in, const int* in_sizes, int n_in,
                              void* d_out, int out_size, void* d_ws, size_t ws_size,
                              hipStream_t stream) {
  const int n = N_NODES_C;
  const int E = in_sizes[0];
  const int B = n / 81;

  const int* src = (const int*)d_in[0];
  const int* dst = (const int*)d_in[1];
  const float* w = (const float*)d_in[2];
  const float* W1 = (const float*)d_in[4];
  const float* b1 = (const float*)d_in[5];
  const float* bias1 = (const float*)d_in[6];
  const float* c1s = (const float*)d_in[7];
  const float* c1p = (const float*)d_in[8];
  const float* c1n = (const float*)d_in[9];
  const float* W2 = (const float*)d_in[10];
  const float* b2 = (const float*)d_in[11];
  const float* bias2 = (const float*)d_in[12];
  const float* c2s = (const float*)d_in[13];
  const float* c2p = (const float*)d_in[14];
  const float* c2n = (const float*)d_in[15];
  const float* k1 = (const float*)d_in[16];
  const float* bk1 = (const float*)d_in[17];
  const float* k2 = (const float*)d_in[18];
  const float* bk2 = (const float*)d_in[19];
  const float* Wfc = (const float*)d_in[20];
  const float* bfc = (const float*)d_in[21];
  const float* Wc = (const float*)d_in[22];
  const float* bc = (const float*)d_in[23];
  float* out = (float*)d_out;

  // workspace layout: 135*n + E floats (~195 MB)
  float* ws = (float*)d_ws;
  float* deg = ws;
  float* mp = ws + (size_t)n;
  float* mn = ws + 2 * (size_t)n;
  float* sp = ws + 3 * (size_t)n;
  float* sn = ws + 4 * (size_t)n;
  float* hp1 = ws + 5 * (size_t)n;
  float* hn1 = ws + 6 * (size_t)n;
  float* coef = ws + 7 * (size_t)n;      // E floats
  float* h1 = coef + (size_t)E;          // 32n
  float* hp2 = h1 + 32 * (size_t)n;      // 32n
  float* hn2 = hp2 + 32 * (size_t)n;     // 32n
  float* h2 = hn2 + 32 * (size_t)n;      // 32n

  const int T = 256;
  k_init_nodes<<<(n + T - 1) / T, T, 0, stream>>>(deg, mp, mn, sp, sn, hp1, hn1, n);
  k_zero_feat<<<(n * HIDC + T - 1) / T, T, 0, stream>>>(hp2, hn2, n * HIDC);
  k_edge_max<<<(E + T - 1) / T, T, 0, stream>>>(w, dst, mp, mn, E);
  k_edge_expsum<<<(E + T - 1) / T, T, 0, stream>>>(w, dst, mp, mn, sp, sn, coef, deg, E);
  k_edge_norm_agg1<<<(E + T - 1) / T, T, 0, stream>>>(w, src, dst, sp, sn, deg, coef, hp1, hn1, E);
  k_node_l1<<<(n * HIDC + T - 1) / T, T, 0, stream>>>(deg, hp1, hn1, W1, b1, bias1, c1s, c1p, c1n,
                                                      h1, n);
  {
    long long tot = (long long)E * 32;
    k_agg2<<<(int)((tot + T - 1) / T), T, 0, stream>>>(src, dst, coef, h1, hp2, hn2, E);
  }
  // one wave per 16 nodes -> 2 threads/node; n % 128 == 0 so EXEC is full
  k_gemm_l2<<<(2 * n) / T, T, 0, stream>>>(h1, hp2, hn2, W2, b2, bias2, c2s, c2p, c2n, h2, n);
  k_head<<<B, 128, 0, stream>>>(h1, h2, k1, bk1, k2, bk2, Wfc, bfc, Wc, bc, out);
  (void)n_in;
  (void)out_size;
  (void)ws_size;
}